// GAU_46797963657716
// MI455X (gfx1250) — compile-verified
//
#include <hip/hip_runtime.h>

typedef __attribute__((ext_vector_type(2))) float v2f;
typedef __attribute__((ext_vector_type(4))) float v4f;
typedef __attribute__((ext_vector_type(8))) float v8f;

#define WMMA_F32_16x16x4(A, Bf, C) \
  __builtin_amdgcn_wmma_f32_16x16x4_f32(false, (A), false, (Bf), (short)0, (C), false, false)

constexpr int B_ = 16, T_ = 288, D_ = 128, O_ = 128, N_ = 307;
constexpr int BT = B_ * T_;        // 4608
constexpr int DO = D_ * O_;        // 16384
constexpr float SIM_SCALE = 0.08838834764831845f;  // 1/sqrt(128)

__device__ __forceinline__ float silu_f(float x) { return x / (1.0f + __expf(-x)); }

// ---------------------------------------------------------------------------
// Kernel 1: per-(b,t) GEMV x·{Wu,Wv,Wz} + silu + q/k scale-shift.
// Memory-bound (0.5 FLOP/byte): stream 906 MB of weights once with NT b128
// loads. One wave per (b,t); lane handles 4 consecutive outputs o.
// ---------------------------------------------------------------------------
__global__ __launch_bounds__(256) void k1_gemv_silu(
    const float* __restrict__ x,
    const float* __restrict__ Wu,
    const float* __restrict__ Wv,
    const float* __restrict__ Wz,
    const float* __restrict__ gamma,
    const float* __restrict__ beta,
    float* __restrict__ qB, float* __restrict__ kB,
    float* __restrict__ vB, float* __restrict__ gB)
{
  __shared__ v4f xs[8][32];
  const int tid  = threadIdx.x;
  const int w    = tid >> 5;
  const int lane = tid & 31;
  const int bt   = blockIdx.x * 8 + w;   // 576 blocks * 8 waves = 4608 exactly

  xs[w][lane] = *(const v4f*)(x + (size_t)bt * D_ + lane * 4);
  // same-wave LDS RAW; compiler inserts s_wait_dscnt

  const size_t base = (size_t)bt * DO + (size_t)(lane * 4);
  const float* pu = Wu + base;
  const float* pv = Wv + base;
  const float* pz = Wz + base;

  v4f au = {0, 0, 0, 0}, av = {0, 0, 0, 0}, az = {0, 0, 0, 0};
  for (int d0 = 0; d0 < 32; ++d0) {
    v4f xv = xs[w][d0];
#pragma unroll
    for (int j = 0; j < 4; ++j) {
      const int off = d0 * 512 + j * 128;  // (4*d0+j)*O
      v4f wu = __builtin_nontemporal_load((const v4f*)(pu + off));
      v4f wv = __builtin_nontemporal_load((const v4f*)(pv + off));
      v4f wz = __builtin_nontemporal_load((const v4f*)(pz + off));
      const float xj = xv[j];
      au += xj * wu;
      av += xj * wv;
      az += xj * wz;
    }
  }

  const int o = lane * 4;
  v4f g0 = *(const v4f*)(gamma + o);
  v4f g1 = *(const v4f*)(gamma + O_ + o);
  v4f b0 = *(const v4f*)(beta + o);
  v4f b1 = *(const v4f*)(beta + O_ + o);
  v4f gate, vs, q, k;
#pragma unroll
  for (int c = 0; c < 4; ++c) {
    gate[c] = silu_f(au[c]);
    vs[c]   = silu_f(av[c]);
    const float zs = silu_f(az[c]);
    q[c] = zs * g0[c] + b0[c];
    k[c] = zs * g1[c] + b1[c];
  }
  const size_t ob = (size_t)bt * O_ + o;
  *(v4f*)(qB + ob) = q;
  *(v4f*)(kB + ob) = k;
  *(v4f*)(vB + ob) = vs;
  *(v4f*)(gB + ob) = gate;
}

// ---------------------------------------------------------------------------
// Kernel 2: attention per (batch, 16-query tile) with f32 WMMA 16x16x4.
// Fragment layout (ISA 7.12.2): A lane L: m=L%16, k=k0+(L<16?0:2), float2.
// B symmetric with n=L%16. C/D: 8 VGPRs, m = r + (L<16?0:8), n = L%16.
// sim kept in LDS (row stride 292 -> 64-bank conflict-free for 4m+k pattern).
// ---------------------------------------------------------------------------
__global__ __launch_bounds__(128) void k2_attention(
    const float* __restrict__ qB, const float* __restrict__ kB,
    const float* __restrict__ vB, const float* __restrict__ gB,
    float* __restrict__ voutB)
{
  __shared__ __align__(16) float sim[16 * 292];
  const int tid  = threadIdx.x;
  const int w    = tid >> 5;
  const int lane = tid & 31;
  const int n16  = lane & 15;
  const int half = lane >> 4;
  const int b    = blockIdx.y;
  const int it   = blockIdx.x;

  // cache q A-fragments in registers (16 rows x 128 K -> 32 float2/lane)
  v2f afrag[32];
  {
    const float* qrow = qB + ((size_t)(b * T_ + it * 16 + n16)) * O_ + half * 2;
#pragma unroll
    for (int kt = 0; kt < 32; ++kt) afrag[kt] = *(const v2f*)(qrow + kt * 4);
  }

  // sim[16 x 288] = q . k^T * 1/sqrt(O); wave w owns key tiles jt = w, w+4, ...
  for (int jt = w; jt < 18; jt += 4) {
    const float* krow = kB + ((size_t)(b * T_ + jt * 16 + n16)) * O_ + half * 2;
    v8f c = {0, 0, 0, 0, 0, 0, 0, 0};
#pragma unroll
    for (int kt = 0; kt < 32; ++kt) {
      v2f bf = *(const v2f*)(krow + kt * 4);
      c = WMMA_F32_16x16x4(afrag[kt], bf, c);
    }
#pragma unroll
    for (int r = 0; r < 8; ++r)
      sim[(r + half * 8) * 292 + jt * 16 + n16] = c[r] * SIM_SCALE;
  }
  __syncthreads();

  // row softmax: 8 threads per row, 36 elements each, shuffle-reduce width 8
  {
    const int row = tid >> 3, j = tid & 7;
    float* rp = sim + row * 292 + j * 36;
    float m = -3.402823466e38f;
#pragma unroll 4
    for (int i = 0; i < 36; ++i) m = fmaxf(m, rp[i]);
    m = fmaxf(m, __shfl_xor(m, 1, 8));
    m = fmaxf(m, __shfl_xor(m, 2, 8));
    m = fmaxf(m, __shfl_xor(m, 4, 8));
    float s = 0.f;
#pragma unroll 4
    for (int i = 0; i < 36; ++i) { const float e = __expf(rp[i] - m); rp[i] = e; s += e; }
    s += __shfl_xor(s, 1, 8);
    s += __shfl_xor(s, 2, 8);
    s += __shfl_xor(s, 4, 8);
    const float inv = 1.f / s;
#pragma unroll 4
    for (int i = 0; i < 36; ++i) rp[i] *= inv;
  }
  __syncthreads();

  // V = A @ v (K = 288), then * gate; wave w owns o-tiles n0 = w, w+4
  const float* srow = sim + n16 * 292 + half * 2;  // A-frag row m = L%16
  for (int n0 = w; n0 < 8; n0 += 4) {
    const int o = n0 * 16 + n16;
    const float* vb = vB + (size_t)b * T_ * O_ + o;
    v8f c = {0, 0, 0, 0, 0, 0, 0, 0};
#pragma unroll 4
    for (int k0 = 0; k0 < 288; k0 += 4) {
      v2f a = *(const v2f*)(srow + k0);
      const int kk = k0 + half * 2;
      v2f bf;
      bf.x = vb[(size_t)kk * O_];
      bf.y = vb[(size_t)(kk + 1) * O_];
      c = WMMA_F32_16x16x4(a, bf, c);
    }
    const float* gp = gB + ((size_t)(b * T_ + it * 16)) * O_ + o;
    float* op       = voutB + ((size_t)(b * T_ + it * 16)) * O_ + o;
#pragma unroll
    for (int r = 0; r < 8; ++r) {
      const int m2 = r + half * 8;
      op[(size_t)m2 * O_] = c[r] * gp[(size_t)m2 * O_];
    }
  }
}

// ---------------------------------------------------------------------------
// Kernel 3: out[b,t,n] = V . W_out^T + b_out, stored transposed as (B,1,N,T).
// N=307 -> 20 n-tiles, last partial: clamp OOB W_out reads, mask stores.
// Per-lane stores are 8 consecutive t values -> two 16B vector stores.
// ---------------------------------------------------------------------------
__global__ __launch_bounds__(128) void k3_proj(
    const float* __restrict__ voutB,
    const float* __restrict__ W_out,
    const float* __restrict__ b_out,
    float* __restrict__ out)
{
  const int tid  = threadIdx.x;
  const int w    = tid >> 5;
  const int lane = tid & 31;
  const int n16  = lane & 15;
  const int half = lane >> 4;
  const int b    = blockIdx.y;
  const int it   = blockIdx.x;

  v2f afrag[32];
  {
    const float* vrow = voutB + ((size_t)(b * T_ + it * 16 + n16)) * O_ + half * 2;
#pragma unroll
    for (int kt = 0; kt < 32; ++kt) afrag[kt] = *(const v2f*)(vrow + kt * 4);
  }

  for (int nt = w; nt < 20; nt += 4) {
    const int rowN = nt * 16 + n16;
    const int rowC = rowN < N_ ? rowN : (N_ - 1);  // clamp: garbage cols never stored
    const float* wr = W_out + (size_t)rowC * O_ + half * 2;
    v8f c = {0, 0, 0, 0, 0, 0, 0, 0};
#pragma unroll
    for (int kt = 0; kt < 32; ++kt) {
      v2f bf = *(const v2f*)(wr + kt * 4);
      c = WMMA_F32_16x16x4(afrag[kt], bf, c);
    }
    if (rowN < N_) {
      const float bo = b_out[rowN];
      float* op = out + ((size_t)b * N_ + rowN) * T_ + it * 16 + half * 8;
      v4f s0 = {c[0] + bo, c[1] + bo, c[2] + bo, c[3] + bo};
      v4f s1 = {c[4] + bo, c[5] + bo, c[6] + bo, c[7] + bo};
      *(v4f*)(op)     = s0;
      *(v4f*)(op + 4) = s1;
    }
  }
}

// ---------------------------------------------------------------------------
extern "C" void kernel_launch(void* const* d_in, const int* in_sizes, int n_in,
                              void* d_out, int out_size, void* d_ws, size_t ws_size,
                              hipStream_t stream) {
  (void)in_sizes; (void)n_in; (void)out_size; (void)ws_size;
  const float* x     = (const float*)d_in[0];
  const float* Wu    = (const float*)d_in[1];
  const float* Wv    = (const float*)d_in[2];
  const float* Wz    = (const float*)d_in[3];
  const float* gamma = (const float*)d_in[4];
  const float* beta  = (const float*)d_in[5];
  const float* Wout  = (const float*)d_in[6];
  const float* bout  = (const float*)d_in[7];
  float* out = (float*)d_out;

  float* ws = (float*)d_ws;
  float* qB = ws;                              // [BT, O]
  float* kB = qB + (size_t)BT * O_;            // [BT, O]
  float* vB = kB + (size_t)BT * O_;            // [BT, O]
  float* gB = vB + (size_t)BT * O_;            // [BT, O]
  float* vo = gB + (size_t)BT * O_;            // [BT, O]  (post-attention V*gate)

  k1_gemv_silu<<<BT / 8, 256, 0, stream>>>(x, Wu, Wv, Wz, gamma, beta, qB, kB, vB, gB);
  k2_attention<<<dim3(T_ / 16, B_), 128, 0, stream>>>(qB, kB, vB, gB, vo);
  k3_proj<<<dim3(T_ / 16, B_), 128, 0, stream>>>(vo, Wout, bout, out);
}